// DGI_86998857548311
// MI455X (gfx1250) — compile-verified
//
#include <hip/hip_runtime.h>
#include <hip/hip_bf16.h>

typedef __attribute__((ext_vector_type(2))) float v2f;
typedef __attribute__((ext_vector_type(8))) float v8f;

#define NF 128   // feature width (in = hidden = 128)

// ---------------------------------------------------------------------------
// Utility kernels
// ---------------------------------------------------------------------------
__global__ void fill_f32(float* __restrict__ p, float v, int n) {
    int i = blockIdx.x * blockDim.x + threadIdx.x;
    if (i < n) p[i] = v;
}

// deg[col[e]] += 1  (self-loop contribution pre-seeded as 1.0 by fill_f32)
__global__ void deg_edges(const int* __restrict__ col, float* __restrict__ deg, int E) {
    int e = blockIdx.x * blockDim.x + threadIdx.x;
    if (e < E) atomicAdd(&deg[col[e]], 1.0f);
}

__global__ void rsqrt_deg(const float* __restrict__ deg, float* __restrict__ dinv, int n) {
    int i = blockIdx.x * blockDim.x + threadIdx.x;
    if (i < n) {
        float d = deg[i];
        dinv[i] = (d > 0.0f) ? rsqrtf(d) : 0.0f;
    }
}

// ---------------------------------------------------------------------------
// Dense GEMM  C[M x 128] = A[M x 128] @ W[128 x 128]   (fp32 WMMA 16x16x4)
// One wave per 16x16 output tile; block = 8 waves covering the 8 N-tiles of
// one 16-row stripe. K loop fully unrolled: 32 x v_wmma_f32_16x16x4_f32.
// ---------------------------------------------------------------------------
__global__ __launch_bounds__(256) void gemm_wmma_f32(
    const float* __restrict__ A,
    const float* __restrict__ W,
    float* __restrict__ C,
    int M)
{
    const int lane  = threadIdx.x & 31;
    const int wave  = threadIdx.x >> 5;       // 0..7 -> N tile
    const int m0    = blockIdx.x * 16;
    const int n0    = wave * 16;
    const int lhalf = lane >> 4;              // 0: K pair {0,1}, 1: K pair {2,3}
    const int l15   = lane & 15;

    int mrow = m0 + l15;                      // A row held by this lane
    if (mrow >= M) mrow = M - 1;              // clamp (keep EXEC all-ones)
    const float* arow = A + (size_t)mrow * NF;

    // keep the 64KB weight matrix hot
    __builtin_prefetch(W + (size_t)n0, 0, 3);

    v8f acc = {};
    #pragma unroll
    for (int k0 = 0; k0 < NF; k0 += 4) {
        const int ka = k0 + 2 * lhalf;
        // A fragment: lane (M=l15) holds K=ka, ka+1   (ISA 32-bit 16x4 layout)
        v2f a = *(const v2f*)(arow + ka);
        // B fragment: lane (N=l15) holds K=ka, ka+1
        v2f b;
        b.x = W[(size_t)ka * NF + n0 + l15];
        b.y = W[(size_t)(ka + 1) * NF + n0 + l15];
        acc = __builtin_amdgcn_wmma_f32_16x16x4_f32(
            /*neg_a=*/false, a, /*neg_b=*/false, b,
            /*c_mod=*/(short)0, acc, /*reuse_a=*/false, /*reuse_b=*/false);
    }

    // C/D layout: lanes 0..15 -> N=lane, VGPR i -> M=i; lanes 16..31 -> M=i+8
    const int nn = n0 + l15;
    #pragma unroll
    for (int i = 0; i < 8; ++i) {
        int m = m0 + i + 8 * lhalf;
        if (m < M) C[(size_t)m * NF + nn] = acc[i];
    }
}

// ---------------------------------------------------------------------------
// out[i][:] = h[i][:] * dinv[i]^2      (self-loop term; also inits buffer)
// one float4 per thread; 32 threads per node row
// ---------------------------------------------------------------------------
__global__ void self_loop_init(const float* __restrict__ h,
                               const float* __restrict__ dinv,
                               float* __restrict__ out, int n)
{
    int idx = blockIdx.x * blockDim.x + threadIdx.x;   // float4 index
    int total = n * (NF / 4);
    if (idx >= total) return;
    int node = idx >> 5;                               // NF/4 == 32 quads/row
    float s = dinv[node];
    s = s * s;
    float4 v = ((const float4*)h)[idx];
    float4 o;
    o.x = v.x * s; o.y = v.y * s; o.z = v.z * s; o.w = v.w * s;
    ((float4*)out)[idx] = o;
}

// ---------------------------------------------------------------------------
// Edge scatter: out[col] += h[row] * dinv[row]*dinv[col]
// One wave per edge; lane l owns features [4l, 4l+4). The h[row] gather is a
// fully-coalesced 512B row; both h and out stay resident in the 192MB L2, so
// the non-returning global_atomic_add_f32 stream runs at L2 bandwidth.
// ---------------------------------------------------------------------------
__global__ __launch_bounds__(256) void edge_aggregate(
    const int* __restrict__ row_idx, const int* __restrict__ col_idx,
    const float* __restrict__ dinv,  const float* __restrict__ h,
    float* __restrict__ out, int E)
{
    int e = blockIdx.x * (blockDim.x >> 5) + (threadIdx.x >> 5);
    if (e >= E) return;
    int lane = threadIdx.x & 31;
    int r = row_idx[e];                 // scalar-uniform within wave: broadcast
    int c = col_idx[e];
    float norm = dinv[r] * dinv[c];

    float4 v = ((const float4*)(h + (size_t)r * NF))[lane];
    float* ob = out + (size_t)c * NF + lane * 4;
    atomicAdd(ob + 0, v.x * norm);
    atomicAdd(ob + 1, v.y * norm);
    atomicAdd(ob + 2, v.z * norm);
    atomicAdd(ob + 3, v.w * norm);
}

// ---------------------------------------------------------------------------
// out = relu(out + bias)
// ---------------------------------------------------------------------------
__global__ void bias_relu(float* __restrict__ out, const float* __restrict__ b, int total) {
    int idx = blockIdx.x * blockDim.x + threadIdx.x;
    if (idx < total) {
        float v = out[idx] + b[idx & (NF - 1)];
        out[idx] = v > 0.0f ? v : 0.0f;
    }
}

// ---------------------------------------------------------------------------
// Host-side launcher
// ---------------------------------------------------------------------------
extern "C" void kernel_launch(void* const* d_in, const int* in_sizes, int n_in,
                              void* d_out, int out_size, void* d_ws, size_t ws_size,
                              hipStream_t stream) {
    const float* x  = (const float*)d_in[0];
    const int*   ei = (const int*)  d_in[1];   // [2, E]: rows then cols
    const float* W1 = (const float*)d_in[2];
    const float* b1 = (const float*)d_in[3];
    const float* W2 = (const float*)d_in[4];
    const float* b2 = (const float*)d_in[5];

    const int n = in_sizes[0] / NF;            // 50000
    const int E = in_sizes[1] / 2;             // 800000
    const int* erow = ei;
    const int* ecol = ei + E;

    // workspace layout (floats)
    float* ws   = (float*)d_ws;
    size_t npad = ((size_t)n + 255) & ~(size_t)255;
    float* deg  = ws;
    float* dinv = ws + npad;
    float* buf0 = dinv + npad;                 // GEMM output
    float* buf1 = buf0 + (size_t)n * NF;       // layer-1 activations
    float* out  = (float*)d_out;

    const int T = 256;
    const int mtiles = (n + 15) / 16;
    const int totalF = n * NF;
    const int quadN  = n * (NF / 4);
    const int edgeBlocks = (E + (T / 32) - 1) / (T / 32);

    // ---- normalization (shared by both layers) ----
    fill_f32<<<(n + T - 1) / T, T, 0, stream>>>(deg, 1.0f, n);            // self-loops
    deg_edges<<<(E + T - 1) / T, T, 0, stream>>>(ecol, deg, E);
    rsqrt_deg<<<(n + T - 1) / T, T, 0, stream>>>(deg, dinv, n);

    // ---- layer 1: relu(GCNConv(x, W1, b1)) -> buf1 ----
    gemm_wmma_f32<<<mtiles, T, 0, stream>>>(x, W1, buf0, n);
    self_loop_init<<<(quadN + T - 1) / T, T, 0, stream>>>(buf0, dinv, buf1, n);
    edge_aggregate<<<edgeBlocks, T, 0, stream>>>(erow, ecol, dinv, buf0, buf1, E);
    bias_relu<<<(totalF + T - 1) / T, T, 0, stream>>>(buf1, b1, totalF);

    // ---- layer 2: relu(GCNConv(buf1, W2, b2)) -> d_out ----
    gemm_wmma_f32<<<mtiles, T, 0, stream>>>(buf1, W2, buf0, n);
    self_loop_init<<<(quadN + T - 1) / T, T, 0, stream>>>(buf0, dinv, out, n);
    edge_aggregate<<<edgeBlocks, T, 0, stream>>>(erow, ecol, dinv, buf0, out, E);
    bias_relu<<<(totalF + T - 1) / T, T, 0, stream>>>(out, b2, totalF);
}